// DistanceEncoder_60241211293812
// MI455X (gfx1250) — compile-verified
//
#include <hip/hip_runtime.h>
#include <math.h>

// ---------------------------------------------------------------------------
// DistanceEncoder for MI455X (gfx1250), fp32 end-to-end via V_WMMA_F32_16X16X4_F32
//   B=8, N=4096, K=16, C_hidden=64
// Pipeline:
//   0) zero stats
//   1) knn_kernel        : WMMA ranking-key (0.5|c|^2 - q.c) via C-init trick,
//                          register top-16 per lane (indices only needed)
//   2) feat_conv1_kernel : build 10-d features, conv1 (10->64) via WMMA, BN1 stats
//   3) bn_finalize       : scale/shift for BN1   (b1 absorbed by training BN)
//   4) bn_act_conv<false>: BN1+lrelu, conv2 via WMMA, BN2 stats
//   5) bn_finalize       : scale/shift for BN2   (b2 absorbed by training BN)
//   6) bn_act_conv<true> : BN2+lrelu, conv3 via WMMA, +b3, scatter to [B,64,N,K]
// ---------------------------------------------------------------------------

typedef __attribute__((ext_vector_type(2))) float v2f;
typedef __attribute__((ext_vector_type(8))) float v8f;

#define B_SZ   8
#define N_PTS  4096
#define K_NN   16
#define NK     (N_PTS * K_NN)      // 65536 = 2^16
#define NPTS   (B_SZ * NK)         // 524288
#define CH     64

static __device__ __forceinline__ v8f wmma_f32_16x16x4(v2f a, v2f b, v8f c) {
  // 8 args: (neg_a, A, neg_b, B, c_mod, C, reuse_a, reuse_b)
  return __builtin_amdgcn_wmma_f32_16x16x4_f32(false, a, false, b, (short)0, c,
                                               false, false);
}

// Intra-wave LDS store->load visibility (cross-lane): explicit CDNA5 DS wait.
static __device__ __forceinline__ void wave_lds_fence() {
  __builtin_amdgcn_wave_barrier();
  asm volatile("s_wait_dscnt 0" ::: "memory");
  __builtin_amdgcn_wave_barrier();
}

// ---------------------------------------------------------------------------
// Kernel 1: KNN. 128 WGs x 256 threads. Each wave owns 32 queries.
// Only neighbor INDICES are needed, so per-row-constant |q|^2 is dropped and
// the ranking key 0.5|c|^2 - q.c is produced entirely inside the WMMA:
//   A = -q (negated once at setup, f32 WMMA has no A-neg), B = c (K=3 pad 0),
//   C initialized per-column to 0.5|c|^2  ->  D = ranking key, no epilogue.
// Top-16 kept per lane in registers (unsorted, replace-max).
// ---------------------------------------------------------------------------
__global__ __launch_bounds__(256) void knn_kernel(const float* __restrict__ xyz,
                                                  int* __restrict__ out_idx) {
  __shared__ float sxyz[3072];      // 1024 candidates AoS (x,y,z)
  __shared__ float schalf[1024];    // 0.5*|c|^2
  __shared__ float stage[8][512];   // per wave: two 16x16 key tiles

  const int tid  = threadIdx.x;
  const int wave = tid >> 5, lane = tid & 31;
  const int hi   = lane >> 4, ln = lane & 15;
  const int wg   = blockIdx.x;            // 128 WGs
  const int b    = wg >> 4;               // / 16
  const int qblk = wg & 15;
  const int q_base = qblk * 256 + wave * 32;
  const float* bx = xyz + (size_t)b * N_PTS * 3;

  // Negated A tiles (16x4, rows=queries): lanes 0-15 hold K=0,1; 16-31 K=2,3
  v2f A1, A2;
  {
    int q1 = q_base + ln;
    float x = bx[q1 * 3], y = bx[q1 * 3 + 1], z = bx[q1 * 3 + 2];
    A1.x = hi ? -z : -x;  A1.y = hi ? 0.f : -y;
    int q2 = q1 + 16;
    x = bx[q2 * 3]; y = bx[q2 * 3 + 1]; z = bx[q2 * 3 + 2];
    A2.x = hi ? -z : -x;  A2.y = hi ? 0.f : -y;
  }

  float bd[16]; int bi[16];
#pragma unroll
  for (int t = 0; t < 16; ++t) { bd[t] = 3.4e38f; bi[t] = t; }
  float curmax = 3.4e38f; int curpos = 0;

  for (int cc = 0; cc < N_PTS; cc += 1024) {
    __syncthreads();
    {
      // 3072 floats, 16B aligned: 768 float4 loads across 256 threads
      const float4* src = (const float4*)(bx + (size_t)cc * 3);
      float4* dst = (float4*)sxyz;
      for (int k = tid; k < 768; k += 256) dst[k] = src[k];
    }
    // prefetch next chunk while we compute (global_prefetch_b8 path)
    if (cc + 1024 < N_PTS)
      __builtin_prefetch(bx + (size_t)(cc + 1024) * 3 + tid * 12, 0, 1);
    __syncthreads();
    for (int i = tid; i < 1024; i += 256) {
      float x = sxyz[i * 3], y = sxyz[i * 3 + 1], z = sxyz[i * 3 + 2];
      schalf[i] = 0.5f * (x * x + y * y + z * z);
    }
    __syncthreads();

    for (int c0 = 0; c0 < 1024; c0 += 16) {
      int cj = c0 + ln;
      float bxv = sxyz[cj * 3 + 2 * hi];     // K0->x (hi=0) / K2->z (hi=1)
      float byv = sxyz[cj * 3 + 1];          // K1->y
      v2f Bt; Bt.x = bxv; Bt.y = hi ? 0.f : byv;   // K3 pad = 0
      float hc = schalf[cj];
      v8f Cin;
#pragma unroll
      for (int v = 0; v < 8; ++v) Cin[v] = hc;     // per-column 0.5|c|^2
      v8f D1 = wmma_f32_16x16x4(A1, Bt, Cin);      // key = 0.5|c|^2 - q.c
      v8f D2 = wmma_f32_16x16x4(A2, Bt, Cin);
#pragma unroll
      for (int v = 0; v < 8; ++v) {
        int m = v + 8 * hi;
        stage[wave][m * 16 + ln]       = D1[v];
        stage[wave][256 + m * 16 + ln] = D2[v];
      }
      wave_lds_fence();
      // lane owns query q_base+lane: hi=0 -> tile1 row ln; hi=1 -> tile2 row ln
      const float* row = &stage[wave][hi * 256 + ln * 16];
      int cbase = cc + c0;
#pragma unroll
      for (int t = 0; t < 16; ++t) {
        float d = row[t];
        if (d < curmax) {
          int ci = cbase + t;
#pragma unroll
          for (int s = 0; s < 16; ++s)
            if (s == curpos) { bd[s] = d; bi[s] = ci; }
          curmax = bd[0]; curpos = 0;
#pragma unroll
          for (int s = 1; s < 16; ++s)
            if (bd[s] > curmax) { curmax = bd[s]; curpos = s; }
        }
      }
      wave_lds_fence();
    }
  }
  int q = q_base + lane;
  int* op = out_idx + ((size_t)b * N_PTS + q) * K_NN;
#pragma unroll
  for (int t = 0; t < 16; ++t) op[t] = bi[t];
}

// ---------------------------------------------------------------------------
// Kernel 2: features + conv1 (10->64, K padded to 12) + BN1 stats.
// 256 thr = 8 waves, 16 points/wave -> 128 points per WG, grid = NPTS/128.
// h1 layout: [64][NPTS] channel-major.  (b1 omitted: cancelled by training BN)
// ---------------------------------------------------------------------------
__global__ __launch_bounds__(256) void feat_conv1_kernel(
    const float* __restrict__ xyz, const int* __restrict__ nn_idx,
    const float* __restrict__ W1, float* __restrict__ h1,
    float* __restrict__ stats) {
  __shared__ float w1s[64][12];
  __shared__ float feat[8][12][16];
  __shared__ float ssum[64], ssq[64];

  const int tid = threadIdx.x;
  const int wave = tid >> 5, lane = tid & 31;
  const int hi = lane >> 4, ln = lane & 15;

  for (int i = tid; i < 64 * 12; i += 256) {
    int r = i / 12, c = i % 12;
    w1s[r][c] = (c < 10) ? W1[r * 10 + c] : 0.0f;
  }
  if (tid < 64) { ssum[tid] = 0.f; ssq[tid] = 0.f; }
  __syncthreads();

  const int p0 = blockIdx.x * 128 + wave * 16;
  if (lane < 16) {
    int p  = p0 + ln;
    int n  = (p >> 4) & (N_PTS - 1);
    int b  = p >> 16;
    const float* bx = xyz + (size_t)b * N_PTS * 3;
    float cx = bx[n * 3], cy = bx[n * 3 + 1], cz = bx[n * 3 + 2];
    int j = nn_idx[p];
    float nx = bx[j * 3], ny = bx[j * 3 + 1], nz = bx[j * 3 + 2];
    float vx = cx - nx, vy = cy - ny, vz = cz - nz;
    float d = sqrtf(fmaxf(vx * vx + vy * vy + vz * vz, 1e-12f));
    feat[wave][0][ln] = cx; feat[wave][1][ln] = cy; feat[wave][2][ln] = cz;
    feat[wave][3][ln] = nx; feat[wave][4][ln] = ny; feat[wave][5][ln] = nz;
    feat[wave][6][ln] = vx; feat[wave][7][ln] = vy; feat[wave][8][ln] = vz;
    feat[wave][9][ln] = d;  feat[wave][10][ln] = 0.f; feat[wave][11][ln] = 0.f;
  }
  wave_lds_fence();

#pragma unroll
  for (int mt = 0; mt < 4; ++mt) {
    v8f acc = {};
#pragma unroll
    for (int k0 = 0; k0 < 12; k0 += 4) {
      v2f a, bt;
      int m = mt * 16 + ln;
      a.x  = w1s[m][k0 + 2 * hi];
      a.y  = w1s[m][k0 + 1 + 2 * hi];
      bt.x = feat[wave][k0 + 2 * hi][ln];
      bt.y = feat[wave][k0 + 1 + 2 * hi][ln];
      acc = wmma_f32_16x16x4(a, bt, acc);
    }
#pragma unroll
    for (int v = 0; v < 8; ++v) {
      int c = mt * 16 + v + 8 * hi;
      float val = acc[v];
      h1[(size_t)c * NPTS + p0 + ln] = val;
      float s = val, q2 = val * val;
#pragma unroll
      for (int mm = 1; mm < 16; mm <<= 1) {
        s += __shfl_xor(s, mm, 32);
        q2 += __shfl_xor(q2, mm, 32);
      }
      if (ln == 0) { atomicAdd(&ssum[c], s); atomicAdd(&ssq[c], q2); }
    }
  }
  __syncthreads();
  if (tid < 64) {
    atomicAdd(&stats[tid], ssum[tid]);
    atomicAdd(&stats[64 + tid], ssq[tid]);
  }
}

// ---------------------------------------------------------------------------
// BN finalize: per-channel scale = g*rsqrt(var+eps), shift = be - mu*scale
// ---------------------------------------------------------------------------
__global__ void bn_finalize_kernel(const float* __restrict__ stats,
                                   const float* __restrict__ g,
                                   const float* __restrict__ be,
                                   float* __restrict__ scale,
                                   float* __restrict__ shift) {
  int c = threadIdx.x;
  if (c < 64) {
    const float inv = 1.0f / (float)NPTS;
    float mu  = stats[c] * inv;
    float var = stats[64 + c] * inv - mu * mu;
    float sc  = g[c] * rsqrtf(var + 1e-5f);
    scale[c] = sc;
    shift[c] = be[c] - mu * sc;
  }
}

// ---------------------------------------------------------------------------
// Kernels 4/6: BN+lrelu on input, 64x64 conv via WMMA (16 K-steps of 4).
// Activation staging uses float4 (global_load_b128 / ds_store_b128).
// FINAL=false: write h_out [64][NPTS] + accumulate stats for next BN.
// FINAL=true : add b3, scatter to output layout [B,64,N,K].
// ---------------------------------------------------------------------------
template <bool FINAL>
__global__ __launch_bounds__(256) void bn_act_conv_kernel(
    const float* __restrict__ hin, const float* __restrict__ Wg,
    const float* __restrict__ scale, const float* __restrict__ shift,
    const float* __restrict__ bias, float* __restrict__ hout,
    float* __restrict__ stats) {
  __shared__ float wsm[64][66];
  __shared__ __align__(16) float xs[8][64][16];
  __shared__ float ssum[64], ssq[64];

  const int tid = threadIdx.x;
  const int wave = tid >> 5, lane = tid & 31;
  const int hi = lane >> 4, ln = lane & 15;

  for (int i = tid; i < 64 * 64; i += 256) wsm[i >> 6][i & 63] = Wg[i];
  if (!FINAL && tid < 64) { ssum[tid] = 0.f; ssq[tid] = 0.f; }
  __syncthreads();

  const int p0 = blockIdx.x * 128 + wave * 16;
  // 64 ch x 16 pts = 256 float4 chunks of 4 pts; 8 iters/lane, b128 traffic
  for (int i = lane; i < 256; i += 32) {
    int c = i >> 2, pt4 = (i & 3) * 4;
    float4 v = *(const float4*)&hin[(size_t)c * NPTS + p0 + pt4];
    float sc = scale[c], sh = shift[c];
    v.x = sc * v.x + sh; v.x = (v.x >= 0.f) ? v.x : 0.01f * v.x;
    v.y = sc * v.y + sh; v.y = (v.y >= 0.f) ? v.y : 0.01f * v.y;
    v.z = sc * v.z + sh; v.z = (v.z >= 0.f) ? v.z : 0.01f * v.z;
    v.w = sc * v.w + sh; v.w = (v.w >= 0.f) ? v.w : 0.01f * v.w;
    *(float4*)&xs[wave][c][pt4] = v;
  }
  wave_lds_fence();

#pragma unroll
  for (int mt = 0; mt < 4; ++mt) {
    v8f acc = {};
#pragma unroll
    for (int k0 = 0; k0 < 64; k0 += 4) {
      v2f a, bt;
      int m = mt * 16 + ln;
      a.x  = wsm[m][k0 + 2 * hi];
      a.y  = wsm[m][k0 + 1 + 2 * hi];
      bt.x = xs[wave][k0 + 2 * hi][ln];
      bt.y = xs[wave][k0 + 1 + 2 * hi][ln];
      acc = wmma_f32_16x16x4(a, bt, acc);
    }
    if (FINAL) {
      int p = p0 + ln;
      int b = p >> 16;
      int r = p & (NK - 1);      // n*K + kk
#pragma unroll
      for (int v = 0; v < 8; ++v) {
        int c = mt * 16 + v + 8 * hi;
        hout[((size_t)b * 64 + c) * NK + r] = acc[v] + bias[c];
      }
    } else {
#pragma unroll
      for (int v = 0; v < 8; ++v) {
        int c = mt * 16 + v + 8 * hi;
        float val = acc[v];
        hout[(size_t)c * NPTS + p0 + ln] = val;
        float s = val, q2 = val * val;
#pragma unroll
        for (int mm = 1; mm < 16; mm <<= 1) {
          s += __shfl_xor(s, mm, 32);
          q2 += __shfl_xor(q2, mm, 32);
        }
        if (ln == 0) { atomicAdd(&ssum[c], s); atomicAdd(&ssq[c], q2); }
      }
    }
  }
  if (!FINAL) {
    __syncthreads();
    if (tid < 64) {
      atomicAdd(&stats[tid], ssum[tid]);
      atomicAdd(&stats[64 + tid], ssq[tid]);
    }
  }
}

__global__ void zero_stats_kernel(float* __restrict__ p, int n) {
  int i = blockIdx.x * blockDim.x + threadIdx.x;
  if (i < n) p[i] = 0.f;
}

// ---------------------------------------------------------------------------
extern "C" void kernel_launch(void* const* d_in, const int* in_sizes, int n_in,
                              void* d_out, int out_size, void* d_ws,
                              size_t ws_size, hipStream_t stream) {
  (void)in_sizes; (void)n_in; (void)out_size; (void)ws_size;
  const float* xyz = (const float*)d_in[0];
  const float* W1  = (const float*)d_in[1];
  /* b1 = d_in[2]: cancelled by training-mode BN */
  const float* g1  = (const float*)d_in[3];
  const float* be1 = (const float*)d_in[4];
  const float* W2  = (const float*)d_in[5];
  /* b2 = d_in[6]: cancelled by training-mode BN */
  const float* g2  = (const float*)d_in[7];
  const float* be2 = (const float*)d_in[8];
  const float* W3  = (const float*)d_in[9];
  const float* b3  = (const float*)d_in[10];

  float* out_f   = (float*)d_out;                           // [B,64,N,K]
  int*   out_idx = (int*)(out_f + (size_t)B_SZ * CH * NK);  // [B,N,K]

  float* wsf    = (float*)d_ws;
  float* stats1 = wsf;                 // 128 (sum | sumsq)
  float* stats2 = wsf + 128;           // 128
  float* sc1 = wsf + 256; float* sh1 = wsf + 320;
  float* sc2 = wsf + 384; float* sh2 = wsf + 448;
  float* h1 = wsf + 1024;                       // [64][NPTS]
  float* h2 = h1 + (size_t)CH * NPTS;           // [64][NPTS]

  zero_stats_kernel<<<1, 256, 0, stream>>>(wsf, 512);
  knn_kernel<<<128, 256, 0, stream>>>(xyz, out_idx);
  feat_conv1_kernel<<<NPTS / 128, 256, 0, stream>>>(xyz, out_idx, W1, h1, stats1);
  bn_finalize_kernel<<<1, 64, 0, stream>>>(stats1, g1, be1, sc1, sh1);
  bn_act_conv_kernel<false><<<NPTS / 128, 256, 0, stream>>>(h1, W2, sc1, sh1,
                                                            nullptr, h2, stats2);
  bn_finalize_kernel<<<1, 64, 0, stream>>>(stats2, g2, be2, sc2, sh2);
  bn_act_conv_kernel<true><<<NPTS / 128, 256, 0, stream>>>(h2, W3, sc2, sh2, b3,
                                                           out_f, nullptr);
}